// TenGCN_25692494365283
// MI455X (gfx1250) — compile-verified
//
#include <hip/hip_runtime.h>
#include <math.h>

// ---------------- dims (match reference) ----------------
#define NNODES 16384      // B*NPG
#define H2DIM  1024       // HID*HID
#define HIDD   32
#define INDIM  128
#define NB     64         // graphs
#define NPGRP  256        // nodes per graph

typedef __attribute__((ext_vector_type(2))) float v2f;
typedef __attribute__((ext_vector_type(8))) float v8f;

// ---------------------------------------------------------
// utility: zero a buffer in float4 chunks
// ---------------------------------------------------------
__global__ void k_zero4(float4* __restrict__ p, int n4) {
    int i = blockIdx.x * blockDim.x + threadIdx.x;
    if (i < n4) p[i] = make_float4(0.f, 0.f, 0.f, 0.f);
}

// ---------------------------------------------------------
// degree count + rsqrt(deg+1)
// ---------------------------------------------------------
__global__ void k_deg(const int* __restrict__ dst, float* __restrict__ deg, int E) {
    int e = blockIdx.x * blockDim.x + threadIdx.x;
    if (e < E) atomicAdd(&deg[dst[e]], 1.0f);
}

__global__ void k_dinv(float* __restrict__ deg, int n) {
    int i = blockIdx.x * blockDim.x + threadIdx.x;
    if (i < n) deg[i] = rsqrtf(deg[i] + 1.0f);
}

// ---------------------------------------------------------
// fp32 WMMA GEMM: C[M,N] = A[M,K] @ B[K,N] (+bias) (+relu)
// Register-blocked: one wave -> 16(M) x (NT*16)(N) strip, NT WMMA
// accumulators sharing one A fragment. NT is compile-time so the
// inner loop is straight-line (no EXEC manipulation). The K-step is
// split load-phase / wmma-phase with an explicit B-fragment array so
// all NT B loads get distinct VGPRs and stay in flight while the
// first WMMAs execute (staged s_wait_loadcnt instead of 0x0 stalls).
//
//   NT==4: block = 8 waves as 2(M) x 4(N) -> 32 x 256 block tile,
//          grid (M/32, N/256). Requires N % 256 == 0.
//   NT==2: block = 8 waves stacked along M -> 128 x 32 block tile,
//          grid (M/128, 1). Requires N == 32.
//
// ISA layouts (CDNA5 ISA, V_WMMA_F32_16X16X4_F32):
//   A 16x4 : lane&15 = M row; (lane>>4)*2 = K base; v2f = {K, K+1}
//   B 4x16 : lane&15 = N col; (lane>>4)*2 = K base; v2f = {K, K+1}
//   C 16x16: lane&15 = N col; VGPR r = row ((lane>>4)*8 + r)
// EXEC is all-ones at every WMMA (no divergent control flow).
// ---------------------------------------------------------
template <int NT, bool BIAS, bool RELU>
__global__ void k_gemm_wmma(const float* __restrict__ A,
                            const float* __restrict__ B,
                            const float* __restrict__ bias,
                            float* __restrict__ C,
                            int M, int N, int K) {
    const int lane = threadIdx.x & 31;
    const int wave = threadIdx.x >> 5;                    // 0..7
    int row0, col0;
    if constexpr (NT == 4) {
        row0 = blockIdx.x * 32 + ((wave >> 2) << 4);
        col0 = blockIdx.y * 256 + ((wave & 3) << 6);
    } else {
        row0 = blockIdx.x * 128 + (wave << 4);
        col0 = 0;
    }

    const int mrow  = lane & 15;
    const int khalf = (lane >> 4) << 1;                   // 0 or 2

    const float* __restrict__ Arow = A + (size_t)(row0 + mrow) * K + khalf;
    const float* __restrict__ Bc   = B + (size_t)khalf * N + (col0 + mrow);

    v8f acc[NT];
#pragma unroll
    for (int t = 0; t < NT; ++t)
        acc[t] = (v8f){0.f, 0.f, 0.f, 0.f, 0.f, 0.f, 0.f, 0.f};

    for (int k = 0; k < K; k += 4) {
        // ---- load phase: A fragment + all NT B fragments ----
        v2f a = *(const v2f*)(Arow + k);                  // A[m, k+khalf..+1]
        size_t r0 = (size_t)k * N;
        size_t r1 = r0 + N;
        v2f bv[NT];
#pragma unroll
        for (int t = 0; t < NT; ++t) {
            bv[t].x = Bc[r0 + (t << 4)];                  // B[k+khalf,   n]
            bv[t].y = Bc[r1 + (t << 4)];                  // B[k+khalf+1, n]
        }
        // ---- wmma phase: consume fragments in issue order ----
#pragma unroll
        for (int t = 0; t < NT; ++t)
            acc[t] = __builtin_amdgcn_wmma_f32_16x16x4_f32(
                false, a, false, bv[t], (short)0, acc[t], false, false);
    }

    const int rbase = row0 + ((lane >> 4) << 3);
#pragma unroll
    for (int t = 0; t < NT; ++t) {
        const int col = col0 + (t << 4) + (lane & 15);
        const float bv = BIAS ? bias[col] : 0.f;
#pragma unroll
        for (int r = 0; r < 8; ++r) {
            float v = acc[t][r] + bv;
            if (RELU) v = fmaxf(v, 0.f);
            C[(size_t)(rbase + r) * N + col] = v;
        }
    }
}

// ---------------------------------------------------------
// edge scatter: agg[dst] += dinv[src]*dinv[dst] * h[src]
// one block (256 thr) per edge, float4 per thread (1024 cols);
// agg fits in the 192MB L2 so the atomics stay on-chip.
// ---------------------------------------------------------
__global__ void k_edge_scatter(const float* __restrict__ h,
                               const int* __restrict__ src,
                               const int* __restrict__ dst,
                               const float* __restrict__ dinv,
                               float* __restrict__ agg) {
    int e = blockIdx.x;
    int s = src[e], d = dst[e];
    float coef = dinv[s] * dinv[d];
    const float4* __restrict__ hs = (const float4*)(h + (size_t)s * H2DIM);
    float* __restrict__ ad = agg + (size_t)d * H2DIM;
    int j = threadIdx.x;                 // 0..255 -> cols 4j..4j+3
    float4 v = hs[j];
    atomicAdd(&ad[4 * j + 0], v.x * coef);
    atomicAdd(&ad[4 * j + 1], v.y * coef);
    atomicAdd(&ad[4 * j + 2], v.z * coef);
    atomicAdd(&ad[4 * j + 3], v.w * coef);
}

// ---------------------------------------------------------
// combine: agg += h * dinv[row]^2 + bias   (self-loop + bias)
// ---------------------------------------------------------
__global__ void k_combine(float* __restrict__ agg, const float* __restrict__ h,
                          const float* __restrict__ dinv, const float* __restrict__ bias) {
    int row = blockIdx.x;
    float d2 = dinv[row];
    d2 *= d2;
    int j = threadIdx.x * 4;
    size_t base = (size_t)row * H2DIM + j;
    float4 a  = *(float4*)(agg + base);
    float4 hv = *(const float4*)(h + base);
    a.x += hv.x * d2 + bias[j + 0];
    a.y += hv.y * d2 + bias[j + 1];
    a.z += hv.z * d2 + bias[j + 2];
    a.w += hv.w * d2 + bias[j + 3];
    *(float4*)(agg + base) = a;
}

// ---------------------------------------------------------
// per-graph mean of h1 and h2 -> S[b][2][1024]
// ---------------------------------------------------------
__global__ void k_mean(const float* __restrict__ h1, const float* __restrict__ h2,
                       float* __restrict__ S) {
    int b = blockIdx.x, j = threadIdx.x;  // 256 threads, 4 cols each
    float s10 = 0, s11 = 0, s12 = 0, s13 = 0;
    float s20 = 0, s21 = 0, s22 = 0, s23 = 0;
    for (int n = 0; n < NPGRP; ++n) {
        size_t base = (size_t)(b * NPGRP + n) * H2DIM + j;
        s10 += h1[base];       s11 += h1[base + 256];
        s12 += h1[base + 512]; s13 += h1[base + 768];
        s20 += h2[base];       s21 += h2[base + 256];
        s22 += h2[base + 512]; s23 += h2[base + 768];
    }
    const float inv = 1.0f / NPGRP;
    float* Sb = S + (size_t)b * 2048;
    Sb[j] = s10 * inv;       Sb[j + 256] = s11 * inv;
    Sb[j + 512] = s12 * inv; Sb[j + 768] = s13 * inv;
    Sb[1024 + j] = s20 * inv;       Sb[1024 + j + 256] = s21 * inv;
    Sb[1024 + j + 512] = s22 * inv; Sb[1024 + j + 768] = s23 * inv;
}

// ---------------------------------------------------------
// conv3x3 stride2 (50->24) + relu + maxpool2x2 (->12x12)
// grid (B, HID); thread = one pooled pixel
// ---------------------------------------------------------
__global__ void k_cnn(const float* __restrict__ PI, const float* __restrict__ cw,
                      const float* __restrict__ cb, float* __restrict__ cf) {
    int b = blockIdx.x, c = blockIdx.y;
    __shared__ float wl[45];
    int tid = threadIdx.x;
    if (tid < 45) wl[tid] = cw[c * 45 + tid];
    __syncthreads();
    if (tid >= 144) return;
    int oy = tid / 12, ox = tid % 12;
    float bc = cb[c];
    float m = -1e30f;
    for (int py = 0; py < 2; ++py)
        for (int px = 0; px < 2; ++px) {
            int iy = (oy * 2 + py) * 2, ix = (ox * 2 + px) * 2;
            float s = bc;
            for (int f = 0; f < 5; ++f)
                for (int ky = 0; ky < 3; ++ky)
                    for (int kx = 0; kx < 3; ++kx)
                        s += PI[(((size_t)b * 5 + f) * 50 + iy + ky) * 50 + ix + kx]
                             * wl[(f * 3 + ky) * 3 + kx];
            s = fmaxf(s, 0.f);
            m = fmaxf(m, s);
        }
    cf[((size_t)(b * 32 + c)) * 144 + oy * 12 + ox] = m;
}

// ---------------------------------------------------------
// fold attend_W into per-channel scalars:
//   scal[0..31] = gamma[c] = sum_x tclp_C[c,x]*attend_W[32+x]
//   scal[32]    = alpha0   = sum_x tclg_A[0,x]*attend_W[x]
//   scal[33]    = alpha1   = sum_x tclg_A[1,x]*attend_W[x]
// ---------------------------------------------------------
__global__ void k_scal(const float* __restrict__ tclgA, const float* __restrict__ tclpC,
                       const float* __restrict__ aw, float* __restrict__ scal) {
    int tid = threadIdx.x;
    if (tid < 32) {
        float g = 0.f;
        for (int x = 0; x < 32; ++x) g += tclpC[tid * 32 + x] * aw[32 + x];
        scal[tid] = g;
    } else if (tid < 34) {
        int a = tid - 32;
        float s = 0.f;
        for (int x = 0; x < 32; ++x) s += tclgA[a * 32 + x] * aw[x];
        scal[tid] = s;
    }
}

// ---------------------------------------------------------
// per-graph finale:
//   Sc = a0*S1 + a1*S2 (32x32);  Pm = tclg_H^T * Sc * tclg_W
//   R[h,w] = sum_c gamma[c]*cfeat[b,c,h,w] (12x12)
//   P2 = tclp_H^T * R * tclp_W
//   att[z*32+y] = Pm[y,z] + P2[y,z] + attend_b   (== flat in ref)
//   out[b,o] = sigmoid(sum_i att[i]*out_W[i,o] + out_b[o])
// ---------------------------------------------------------
__global__ void k_finale(const float* __restrict__ S, const float* __restrict__ cfeat,
                         const float* __restrict__ scal,
                         const float* __restrict__ tH, const float* __restrict__ tW,
                         const float* __restrict__ pH, const float* __restrict__ pW,
                         const float* __restrict__ attb,
                         const float* __restrict__ outW, const float* __restrict__ outb,
                         float* __restrict__ out) {
    __shared__ float Sc[1024], tmp[1024], Pm[1024], R[144], T2[384], att[1024], red[256];
    int b = blockIdx.x, tid = threadIdx.x;
    float a0 = scal[32], a1 = scal[33];
    const float* S1 = S + (size_t)b * 2048;
    const float* S2 = S1 + 1024;
    for (int i = tid; i < 1024; i += 256) Sc[i] = a0 * S1[i] + a1 * S2[i];
    __syncthreads();
    for (int i = tid; i < 1024; i += 256) {     // tmp[h,z] = sum_w Sc[h,w]*tW[w,z]
        int h = i >> 5, z = i & 31;
        float s = 0.f;
        for (int w = 0; w < 32; ++w) s += Sc[h * 32 + w] * tW[w * 32 + z];
        tmp[i] = s;
    }
    __syncthreads();
    for (int i = tid; i < 1024; i += 256) {     // Pm[y,z] = sum_h tH[h,y]*tmp[h,z]
        int y = i >> 5, z = i & 31;
        float s = 0.f;
        for (int h = 0; h < 32; ++h) s += tH[h * 32 + y] * tmp[h * 32 + z];
        Pm[i] = s;
    }
    if (tid < 144) {                            // R[h,w] = sum_c gamma[c]*cfeat
        float s = 0.f;
        for (int c = 0; c < 32; ++c) s += scal[c] * cfeat[(size_t)(b * 32 + c) * 144 + tid];
        R[tid] = s;
    }
    __syncthreads();
    for (int i = tid; i < 384; i += 256) {      // T2[h,z] = sum_w R[h,w]*pW[w,z]
        int h = i >> 5, z = i & 31;
        float s = 0.f;
        for (int w = 0; w < 12; ++w) s += R[h * 12 + w] * pW[w * 32 + z];
        T2[i] = s;
    }
    __syncthreads();
    float ab = attb[0];
    for (int i = tid; i < 1024; i += 256) {     // att[z*32+y]
        int z = i >> 5, y = i & 31;
        float s = Pm[y * 32 + z];
        for (int h = 0; h < 12; ++h) s += pH[h * 32 + y] * T2[h * 32 + z];
        att[i] = s + ab;
    }
    __syncthreads();
    int o = tid & 7;                            // out = sigmoid(att @ out_W + out_b)
    float p = 0.f;
    for (int i = tid >> 3; i < 1024; i += 32) p += att[i] * outW[i * 8 + o];
    red[tid] = p;
    __syncthreads();
    for (int s = 128; s >= 8; s >>= 1) {
        if (tid < s) red[tid] += red[tid + s];
        __syncthreads();
    }
    if (tid < 8) {
        float v = red[tid] + outb[tid];
        out[b * 8 + tid] = 1.0f / (1.0f + expf(-v));
    }
}

// ---------------------------------------------------------
extern "C" void kernel_launch(void* const* d_in, const int* in_sizes, int n_in,
                              void* d_out, int out_size, void* d_ws, size_t ws_size,
                              hipStream_t stream) {
    const float* x     = (const float*)d_in[0];
    const int*   ei    = (const int*)d_in[1];
    const float* PI    = (const float*)d_in[2];
    const float* gW0   = (const float*)d_in[3];
    const float* gb0   = (const float*)d_in[4];
    const float* gW1   = (const float*)d_in[5];
    const float* gb1   = (const float*)d_in[6];
    const float* m0W1  = (const float*)d_in[7];
    const float* m0b1  = (const float*)d_in[8];
    const float* m0W2  = (const float*)d_in[9];
    const float* m0b2  = (const float*)d_in[10];
    const float* m1W1  = (const float*)d_in[11];
    const float* m1b1  = (const float*)d_in[12];
    const float* m1W2  = (const float*)d_in[13];
    const float* m1b2  = (const float*)d_in[14];
    const float* tclgA = (const float*)d_in[15];
    const float* tclgH = (const float*)d_in[16];
    const float* tclgW = (const float*)d_in[17];
    const float* convw = (const float*)d_in[18];
    const float* convb = (const float*)d_in[19];
    const float* tclpC = (const float*)d_in[20];
    const float* tclpH = (const float*)d_in[21];
    const float* tclpW = (const float*)d_in[22];
    const float* attW  = (const float*)d_in[23];
    const float* attb  = (const float*)d_in[24];
    const float* outW  = (const float*)d_in[25];
    const float* outb  = (const float*)d_in[26];
    float* out = (float*)d_out;

    const int E = in_sizes[1] / 2;
    const int* src = ei;
    const int* dst = ei + E;

    // workspace layout (floats)
    float* ws = (float*)d_ws;
    const size_t NH = (size_t)NNODES * H2DIM;          // 16.7M floats
    float* buf0  = ws;                                  // GEMM out / h2
    float* buf1  = buf0 + NH;                           // agg / gcn out
    float* buf2  = buf1 + NH;                           // h1
    float* tbuf  = buf2 + NH;                           // MLP bottleneck [N,32]
    float* dinv  = tbuf + (size_t)NNODES * HIDD;        // 16384
    float* Smean = dinv + NNODES;                       // 64*2048
    float* cfeat = Smean + (size_t)NB * 2 * H2DIM;      // 64*32*144
    float* scal  = cfeat + (size_t)NB * HIDD * 144;     // 34

    // degrees -> rsqrt(deg+1)
    k_zero4<<<(NNODES / 4 + 255) / 256, 256, 0, stream>>>((float4*)dinv, NNODES / 4);
    k_deg<<<(E + 255) / 256, 256, 0, stream>>>(dst, dinv, E);
    k_dinv<<<(NNODES + 255) / 256, 256, 0, stream>>>(dinv, NNODES);

    dim3 gBig(NNODES / 32, H2DIM / 256);   // 512 x 4 blocks, N=1024, NT=4
    dim3 gSml(NNODES / 128, 1);            // 128 blocks,      N=32,   NT=2
    const int zeroBlocks = (int)(NH / 4 / 256);

    // ---- GCN layer 1 + MLP0 ----
    k_gemm_wmma<4, false, false><<<gBig, 256, 0, stream>>>(x, gW0, nullptr, buf0, NNODES, H2DIM, INDIM);
    k_zero4<<<zeroBlocks, 256, 0, stream>>>((float4*)buf1, (int)(NH / 4));
    k_edge_scatter<<<E, 256, 0, stream>>>(buf0, src, dst, dinv, buf1);
    k_combine<<<NNODES, 256, 0, stream>>>(buf1, buf0, dinv, gb0);
    k_gemm_wmma<2, true, true><<<gSml, 256, 0, stream>>>(buf1, m0W1, m0b1, tbuf, NNODES, HIDD, H2DIM);
    k_gemm_wmma<4, true, false><<<gBig, 256, 0, stream>>>(tbuf, m0W2, m0b2, buf2, NNODES, H2DIM, HIDD);

    // ---- GCN layer 2 + MLP1 ----
    k_gemm_wmma<4, false, false><<<gBig, 256, 0, stream>>>(buf2, gW1, nullptr, buf0, NNODES, H2DIM, H2DIM);
    k_zero4<<<zeroBlocks, 256, 0, stream>>>((float4*)buf1, (int)(NH / 4));
    k_edge_scatter<<<E, 256, 0, stream>>>(buf0, src, dst, dinv, buf1);
    k_combine<<<NNODES, 256, 0, stream>>>(buf1, buf0, dinv, gb1);
    k_gemm_wmma<2, true, true><<<gSml, 256, 0, stream>>>(buf1, m1W1, m1b1, tbuf, NNODES, HIDD, H2DIM);
    k_gemm_wmma<4, true, false><<<gBig, 256, 0, stream>>>(tbuf, m1W2, m1b2, buf0, NNODES, H2DIM, HIDD);

    // ---- back end (algebraically folded) ----
    k_mean<<<NB, 256, 0, stream>>>(buf2, buf0, Smean);
    k_cnn<<<dim3(NB, HIDD), 256, 0, stream>>>(PI, convw, convb, cfeat);
    k_scal<<<1, 64, 0, stream>>>(tclgA, tclpC, attW, scal);
    k_finale<<<NB, 256, 0, stream>>>(Smean, cfeat, scal, tclgH, tclgW, tclpH, tclpW,
                                     attb, outW, outb, out);
}